// TMSCNNConvolutionGPU_74294344286540
// MI455X (gfx1250) — compile-verified
//
#include <hip/hip_runtime.h>

typedef float v2f __attribute__((ext_vector_type(2)));
typedef float v4f __attribute__((ext_vector_type(4)));
typedef float v8f __attribute__((ext_vector_type(8)));

#define N_GROUPS   100000
#define F_IN       64
#define F_OUT      64
#define KBANKS     9
#define TERMS      36

#define WAVES            8
#define GROUPS_PER_WAVE  32                          // two 16-row M-tiles per wave
#define GROUPS_PER_BLOCK (WAVES * GROUPS_PER_WAVE)   // 256

#define WMT_STRIDE 580   // 576 + 4 pad floats -> 4-bank skew per row, conflict-free b64 reads
#define Y_STRIDE   68    // 64 + 4 pad floats

__global__ __launch_bounds__(256)
void tmscnn_fused_kernel(const float* __restrict__ x,
                         const float* __restrict__ scaling,
                         const float* __restrict__ w,
                         const float* __restrict__ bias,
                         const int*   __restrict__ idx,
                         float*       __restrict__ out)
{
    // W^T in LDS: WmT[o][k*64+f] = w[k][o][f]   (64 rows x 580 floats, 148480 B)
    __shared__ float WmT[F_OUT * WMT_STRIDE];
    // Per-wave gathered/scaled A tiles: 32 groups x 64 f, padded (8704 B/wave, 69632 B)
    __shared__ float Ybuf[WAVES][GROUPS_PER_WAVE * Y_STRIDE];

    const int tid = threadIdx.x;

    // ---- Stage 0: cooperative load of W^T into LDS (9216 float4 = 36/thread) ----
    for (int i = tid; i < (KBANKS * F_OUT * F_IN) / 4; i += 256) {
        int k   = i >> 10;          // 1024 float4 per k-bank
        int rem = i & 1023;
        int o   = rem >> 4;
        int fv  = rem & 15;
        v4f val = *(const v4f*)(w + (size_t)i * 4);
        *(v4f*)(&WmT[o * WMT_STRIDE + k * F_IN + fv * 4]) = val;
    }
    __syncthreads();

    const int wave = tid >> 5;
    const int lane = tid & 31;
    const int half = lane >> 4;   // h: upper half-wave flag
    const int l16  = lane & 15;

    const int g0 = blockIdx.x * GROUPS_PER_BLOCK + wave * GROUPS_PER_WAVE;
    float* yw = &Ybuf[wave][0];

    // Eight 16x16 f32 accumulators: [mtile 0..1][otile 0..3]
    v8f a00 = {}, a01 = {}, a02 = {}, a03 = {};
    v8f a10 = {}, a11 = {}, a12 = {}, a13 = {};

    for (int k = 0; k < KBANKS; ++k) {
        // ---- Gather + scale: y[m][f] = sum_{t4} s * x[idx][f], half-wave per row ----
        for (int i = 0; i < 16; ++i) {
            int m  = 2 * i + half;                    // 0..31
            int g  = g0 + m;
            int gc = (g < N_GROUPS) ? g : 0;          // clamp; tail stores are guarded
            v4f acc = {};
            #pragma unroll
            for (int t4 = 0; t4 < 4; ++t4) {
                int   t   = 4 * k + t4;
                int   row = idx[gc * TERMS + t];      // half-wave-uniform -> coalesced
                float s   = scaling[gc * TERMS + t];
                v4f xv = *(const v4f*)(x + (size_t)row * F_IN + l16 * 4);
                acc += s * xv;
            }
            *(v4f*)(&yw[m * Y_STRIDE + l16 * 4]) = acc;
        }
        asm volatile("s_wait_dscnt 0" ::: "memory");  // cross-lane LDS RAW fence

        // ---- WMMA: 16 K-chunks of 4; 2 M-tiles x 4 N-tiles = 8 WMMA per chunk ----
        #pragma unroll 2
        for (int c = 0; c < 16; ++c) {
            int f0 = 4 * c + 2 * half;                // A/B frag: lane holds K = 2h+v
            v2f aL = *(const v2f*)(&yw[(l16)      * Y_STRIDE + f0]);   // rows 0..15
            v2f aH = *(const v2f*)(&yw[(16 + l16) * Y_STRIDE + f0]);   // rows 16..31
            int kf = k * F_IN + f0;
            v2f b0 = *(const v2f*)(&WmT[(0 * 16 + l16) * WMT_STRIDE + kf]);
            v2f b1 = *(const v2f*)(&WmT[(1 * 16 + l16) * WMT_STRIDE + kf]);
            v2f b2 = *(const v2f*)(&WmT[(2 * 16 + l16) * WMT_STRIDE + kf]);
            v2f b3 = *(const v2f*)(&WmT[(3 * 16 + l16) * WMT_STRIDE + kf]);
            a00 = __builtin_amdgcn_wmma_f32_16x16x4_f32(false, aL, false, b0, (short)0, a00, false, false);
            a01 = __builtin_amdgcn_wmma_f32_16x16x4_f32(false, aL, false, b1, (short)0, a01, false, false);
            a02 = __builtin_amdgcn_wmma_f32_16x16x4_f32(false, aL, false, b2, (short)0, a02, false, false);
            a03 = __builtin_amdgcn_wmma_f32_16x16x4_f32(false, aL, false, b3, (short)0, a03, false, false);
            a10 = __builtin_amdgcn_wmma_f32_16x16x4_f32(false, aH, false, b0, (short)0, a10, false, false);
            a11 = __builtin_amdgcn_wmma_f32_16x16x4_f32(false, aH, false, b1, (short)0, a11, false, false);
            a12 = __builtin_amdgcn_wmma_f32_16x16x4_f32(false, aH, false, b2, (short)0, a12, false, false);
            a13 = __builtin_amdgcn_wmma_f32_16x16x4_f32(false, aH, false, b3, (short)0, a13, false, false);
        }
        asm volatile("s_wait_dscnt 0" ::: "memory");  // A-frag reads done before Y reuse
    }

    // ---- Epilogue: bias + ReLU + store (D: vgpr r -> M = mtile*16 + 8*half + r) ----
    float bb0 = bias[0 * 16 + l16];
    float bb1 = bias[1 * 16 + l16];
    float bb2 = bias[2 * 16 + l16];
    float bb3 = bias[3 * 16 + l16];
    #pragma unroll
    for (int r = 0; r < 8; ++r) {
        int gL = g0 + 8 * half + r;          // M-tile 0
        if (gL < N_GROUPS) {
            float* orow = out + (size_t)gL * F_OUT + l16;
            orow[0]  = fmaxf(a00[r] + bb0, 0.f);
            orow[16] = fmaxf(a01[r] + bb1, 0.f);
            orow[32] = fmaxf(a02[r] + bb2, 0.f);
            orow[48] = fmaxf(a03[r] + bb3, 0.f);
        }
        int gH = g0 + 16 + 8 * half + r;     // M-tile 1
        if (gH < N_GROUPS) {
            float* orow = out + (size_t)gH * F_OUT + l16;
            orow[0]  = fmaxf(a10[r] + bb0, 0.f);
            orow[16] = fmaxf(a11[r] + bb1, 0.f);
            orow[32] = fmaxf(a12[r] + bb2, 0.f);
            orow[48] = fmaxf(a13[r] + bb3, 0.f);
        }
    }
}

extern "C" void kernel_launch(void* const* d_in, const int* in_sizes, int n_in,
                              void* d_out, int out_size, void* d_ws, size_t ws_size,
                              hipStream_t stream) {
    const float* x       = (const float*)d_in[0];
    const float* scaling = (const float*)d_in[1];
    const float* w       = (const float*)d_in[2];
    const float* b       = (const float*)d_in[3];
    const int*   idx     = (const int*)  d_in[4];
    float*       out     = (float*)d_out;

    dim3 grid((N_GROUPS + GROUPS_PER_BLOCK - 1) / GROUPS_PER_BLOCK);
    tmscnn_fused_kernel<<<grid, dim3(256), 0, stream>>>(x, scaling, w, b, idx, out);
}